// SeqSelfAttention_37666863186142
// MI455X (gfx1250) — compile-verified
//
#include <hip/hip_runtime.h>
#include <hip/hip_bf16.h>

// Problem constants (from reference): B=2, L=1024, D=256, U=128, window=128.
#define LSEQ 1024
#define DDIM 256
#define UDIM 128
#define HALFW 64
#define WPAD 144   // padded banded-key range per 16-query tile (>= 143), /4 and /16 aligned

typedef __attribute__((ext_vector_type(2))) float v2f;
typedef __attribute__((ext_vector_type(8))) float v8f;

#if defined(__AMDGCN__) && __has_builtin(__builtin_amdgcn_global_load_async_to_lds_b128) && \
    __has_builtin(__builtin_amdgcn_s_wait_asynccnt)
#define USE_ASYNC_LDS 1
// Exact pointee type from hipcc's diagnostic: int __attribute__((vector_size(16))).
typedef int async_v4i __attribute__((vector_size(4 * sizeof(int))));
typedef __attribute__((address_space(1))) async_v4i* g_v4i_ptr;   // global src
typedef __attribute__((address_space(3))) async_v4i* l_v4i_ptr;   // LDS dst
#else
#define USE_ASYNC_LDS 0
#endif

__device__ __forceinline__ float tanh_fast(float t) {
    t = fminf(fmaxf(t, -15.0f), 15.0f);
    const float ex = __expf(2.0f * t);              // v_exp_f32 path
    return __fdividef(ex - 1.0f, ex + 1.0f);        // fast rcp, no IEEE div
}

// ---------------------------------------------------------------------------
// Kernel A: Q = x @ Wt + bh, K = x @ Wx   via V_WMMA_F32_16X16X4_F32
// grid = (BL/16, U/16, 2), block = 32 (one wave)
// ---------------------------------------------------------------------------
__global__ __launch_bounds__(32) void proj_kernel(
    const float* __restrict__ x,   // (B*L, D) row-major
    const float* __restrict__ Wt,  // (D, U)
    const float* __restrict__ Wx,  // (D, U)
    const float* __restrict__ bh,  // (U)
    float* __restrict__ Qb,        // (B*L, U)
    float* __restrict__ Kp)        // (B*L, U)
{
    const int mt    = blockIdx.x;          // M tile over B*L
    const int nt    = blockIdx.y;          // N tile over U
    const int which = blockIdx.z;          // 0 -> Q (+bh), 1 -> K
    const float* Wm = which ? Wx : Wt;
    float* Out      = which ? Kp : Qb;

    const int lane = threadIdx.x;          // 0..31
    const int r    = lane & 15;            // row / col within tile
    const int h    = lane >> 4;            // half-wave selects K pair
    const int m0   = mt * 16;
    const int n0   = nt * 16;

    const float* xrow = x + (size_t)(m0 + r) * DDIM;

    v8f c = {};
    for (int k0 = 0; k0 < DDIM; k0 += 4) {
        // A: 16x4 f32 tile. VGPR0 = K {0,2}, VGPR1 = K {1,3} per half-wave.
        v2f a;
        a.x = xrow[k0 + 2 * h];
        a.y = xrow[k0 + 2 * h + 1];
        // B: 4x16 f32 tile, rows striped across lanes within a VGPR.
        v2f bb;
        bb.x = Wm[(size_t)(k0 + 2 * h) * UDIM + n0 + r];
        bb.y = Wm[(size_t)(k0 + 2 * h + 1) * UDIM + n0 + r];
        c = __builtin_amdgcn_wmma_f32_16x16x4_f32(false, a, false, bb,
                                                  (short)0, c, false, false);
    }

    const float bias = which ? 0.0f : bh[n0 + r];
    #pragma unroll
    for (int j = 0; j < 8; ++j) {
        // C layout: lane<16 -> M=j, lane>=16 -> M=j+8; N = lane&15.
        Out[(size_t)(m0 + j + 8 * h) * UDIM + n0 + r] = c[j] + bias;
    }
}

// ---------------------------------------------------------------------------
// Kernel B: banded scores + softmax + P @ Xwindow (WMMA f32)
// grid = (L/16, B), block = 256 (8 waves; 2 output N-tiles per wave)
// ---------------------------------------------------------------------------
__global__ __launch_bounds__(256) void attn_kernel(
    const float* __restrict__ x,   // (B, L, D)
    const float* __restrict__ Wa,  // (U)
    const float* __restrict__ Qb,  // (B*L, U)  already has +bh
    const float* __restrict__ Kp,  // (B*L, U)
    float* __restrict__ out)       // (B, L, D)
{
    __shared__ __align__(16) float sQ[16 * UDIM];
    __shared__ __align__(16) float sWa[UDIM];
    __shared__ __align__(16) float sP[16 * WPAD];

    const int qt  = blockIdx.x;
    const int b   = blockIdx.y;
    const int q0  = qt * 16;
    const int klo = (q0 - HALFW > 0) ? (q0 - HALFW) : 0;
    const int tid = threadIdx.x;

    // ---- Stage query tile + Wa into LDS (async data-mover path on CDNA5) --
    const float* Qsrc = Qb + (size_t)(b * LSEQ + q0) * UDIM;  // 16x128 contiguous
#if USE_ASYNC_LDS
    {
        // 2048 floats = 512 x b128; 256 lanes x 2 transfers. ASYNCcnt-tracked.
        #pragma unroll
        for (int j = 0; j < 2; ++j) {
            const int i = tid + j * 256;
            __builtin_amdgcn_global_load_async_to_lds_b128(
                (g_v4i_ptr)(const void*)(Qsrc + 4 * i),
                (l_v4i_ptr)(void*)&sQ[4 * i], 0, 0);
        }
        if (tid < UDIM / 4) {
            __builtin_amdgcn_global_load_async_to_lds_b128(
                (g_v4i_ptr)(const void*)(Wa + 4 * tid),
                (l_v4i_ptr)(void*)&sWa[4 * tid], 0, 0);
        }
        __builtin_amdgcn_s_wait_asynccnt(0);
    }
#else
    for (int i = tid; i < 16 * UDIM; i += 256) sQ[i] = Qsrc[i];
    if (tid < UDIM) sWa[tid] = Wa[tid];
#endif
    __syncthreads();

    // ---- Banded emission scores: e[q][j] = Wa . tanh(qvec + kvec) ----
    for (int p = tid; p < 16 * WPAD; p += 256) {
        const int qq = p / WPAD;
        const int j  = p - qq * WPAD;
        const int kk = klo + j;
        const int q  = q0 + qq;
        float e = -1.0e30f;
        if (kk < LSEQ && kk >= q - HALFW && kk < q + HALFW) {
            const float4* __restrict__ k4 =
                (const float4*)(Kp + (size_t)(b * LSEQ + kk) * UDIM);
            const float4* q4 = (const float4*)(sQ + qq * UDIM);
            const float4* w4 = (const float4*)sWa;
            float acc = 0.0f;
            #pragma unroll 8
            for (int u = 0; u < UDIM / 4; ++u) {
                const float4 qv = q4[u];   // ds_load_b128
                const float4 kv = k4[u];   // global_load_b128
                const float4 wv = w4[u];
                acc = fmaf(wv.x, tanh_fast(qv.x + kv.x), acc);
                acc = fmaf(wv.y, tanh_fast(qv.y + kv.y), acc);
                acc = fmaf(wv.z, tanh_fast(qv.z + kv.z), acc);
                acc = fmaf(wv.w, tanh_fast(qv.w + kv.w), acc);
            }
            e = acc;   // +ba dropped: softmax-invariant constant
        }
        sP[qq * WPAD + j] = e;
    }
    __syncthreads();

    // ---- Row softmax (padded / masked entries underflow to exactly 0) ----
    if (tid < 16) {
        float* row = sP + tid * WPAD;
        float m = -1.0e30f;
        for (int j = 0; j < WPAD; ++j) m = fmaxf(m, row[j]);
        float s = 0.0f;
        for (int j = 0; j < WPAD; ++j) {
            const float pe = __expf(row[j] - m);
            row[j] = pe;
            s += pe;
        }
        const float inv = __fdividef(1.0f, s);
        for (int j = 0; j < WPAD; ++j) row[j] *= inv;
    }
    __syncthreads();

    // ---- v = P(16 x WPAD) @ Xwin(WPAD x 256), f32 WMMA, 2 N-tiles/wave ----
    const int wave = tid >> 5;             // 0..7
    const int lane = tid & 31;
    const int r    = lane & 15;
    const int hh   = lane >> 4;
    const int n0   = wave * 32;            // 8 waves x 32 cols = 256 = D

    v8f c0 = {}, c1 = {};
    for (int k0 = 0; k0 < WPAD; k0 += 4) {
        const int kr = k0 + 2 * hh;
        v2f a;
        a.x = sP[r * WPAD + kr];
        a.y = sP[r * WPAD + kr + 1];
        // Clamp key index: clamped rows have P == 0, contribution is exact 0.
        const int kk0 = (klo + kr     < LSEQ) ? (klo + kr)     : (LSEQ - 1);
        const int kk1 = (klo + kr + 1 < LSEQ) ? (klo + kr + 1) : (LSEQ - 1);
        const float* x0 = x + (size_t)(b * LSEQ + kk0) * DDIM;
        const float* x1 = x + (size_t)(b * LSEQ + kk1) * DDIM;
        v2f b0; b0.x = x0[n0 +  0 + r]; b0.y = x1[n0 +  0 + r];
        v2f b1; b1.x = x0[n0 + 16 + r]; b1.y = x1[n0 + 16 + r];
        c0 = __builtin_amdgcn_wmma_f32_16x16x4_f32(false, a, false, b0, (short)0, c0, false, false);
        c1 = __builtin_amdgcn_wmma_f32_16x16x4_f32(false, a, false, b1, (short)0, c1, false, false);
    }

    float* orow = out + (size_t)(b * LSEQ + q0) * DDIM;
    #pragma unroll
    for (int j = 0; j < 8; ++j) {
        const int rr = j + 8 * hh;
        orow[(size_t)rr * DDIM + n0 +  0 + r] = c0[j];
        orow[(size_t)rr * DDIM + n0 + 16 + r] = c1[j];
    }
}

// ---------------------------------------------------------------------------
extern "C" void kernel_launch(void* const* d_in, const int* in_sizes, int n_in,
                              void* d_out, int out_size, void* d_ws, size_t ws_size,
                              hipStream_t stream) {
    (void)in_sizes; (void)n_in; (void)out_size; (void)ws_size;
    const float* x  = (const float*)d_in[0];
    const float* Wt = (const float*)d_in[1];
    const float* Wx = (const float*)d_in[2];
    const float* Wa = (const float*)d_in[3];
    const float* bh = (const float*)d_in[4];
    // d_in[5] = ba: constant shift before softmax -> mathematically a no-op.

    float* out = (float*)d_out;
    float* Qb  = (float*)d_ws;                        // (B*L, U) = 1 MB
    float* Kp  = Qb + (size_t)2 * LSEQ * UDIM;        // (B*L, U) = 1 MB

    // Projections: 2048 rows -> 128 M-tiles; 128 cols -> 8 N-tiles; z: Q/K.
    proj_kernel<<<dim3(128, 8, 2), 32, 0, stream>>>(x, Wt, Wx, bh, Qb, Kp);
    // Attention: one block per (16-query tile, batch), 8 waves.
    attn_kernel<<<dim3(LSEQ / 16, 2), 256, 0, stream>>>(x, Wa, Qb, Kp, out);
}